// NormalizePixelPts_7722351198856
// MI455X (gfx1250) — compile-verified
//
#include <hip/hip_runtime.h>
#include <hip/hip_bf16.h>

// ---------------------------------------------------------------------------
// NormalizePixelPts: out[f, c] = in[f, c] / div(c), div(c) in {1, 1280, 720}
// HBM-roofline streaming kernel (AI = 1 flop / 8 B -> memory bound, ~49 us
// floor at 23.3 TB/s). Flat grid-stride loop over 32 B (2 x float4) units,
// non-temporal b128 loads/stores (stream > L2), per-block LDS LUT of
// reciprocal multipliers, incremental column tracking (no per-iter modulo),
// unguarded speculative global_prefetch_b8 pipelining.
// ---------------------------------------------------------------------------

#define FEAT 4337u
typedef __attribute__((ext_vector_type(4))) float f32x4;

// Reciprocal multiplier for column c, derived from the reference index walk:
//   c==2 -> 1/W, c==3 -> 1/H, c in [9,96]: odd->1/W even->1/H
//   blocks at b = 97 + k*2120 (k=0,1):
//     o = c-b in [0,160):   even->1/W odd->1/H
//     o in [160,360):       1
//     o in [360,2120):      (o-360) even->1/W odd->1/H
__device__ __forceinline__ float col_mult(unsigned c) {
    const float RW = 1.0f / 1280.0f;
    const float RH = 1.0f / 720.0f;
    float r = 1.0f;
    if (c >= 97u) {
        unsigned o = c - 97u;
        if (o >= 2120u) o -= 2120u;
        if (o < 160u) {
            r = (o & 1u) ? RH : RW;
        } else if (o >= 360u) {
            r = ((o - 360u) & 1u) ? RH : RW;
        } // [160,360) stays 1.0
    } else {
        if (c == 2u)      r = RW;
        else if (c == 3u) r = RH;
        else if (c >= 9u) r = (c & 1u) ? RW : RH;
    }
    return r;
}

__global__ void __launch_bounds__(256)
normalize_pixel_pts_kernel(const float* __restrict__ in,
                           float* __restrict__ out,
                           unsigned long long npair, // number of 8-element units
                           unsigned long long n)     // total elements
{
    // --- Build reciprocal LUT in LDS (17.3 KB of the 320 KB WGP pool) ---
    __shared__ float mlut[FEAT];
    for (unsigned c = threadIdx.x; c < FEAT; c += blockDim.x)
        mlut[c] = col_mult(c);
    __syncthreads();

    const unsigned long long tid    = (unsigned long long)blockIdx.x * blockDim.x + threadIdx.x;
    const unsigned long long stride = (unsigned long long)gridDim.x * blockDim.x;

    const f32x4* __restrict__ in4  = (const f32x4*)in;
    f32x4* __restrict__       out4 = (f32x4*)out;

    // Column of element 8*tid, maintained incrementally mod FEAT.
    unsigned c     = (unsigned)((tid * 8ull) % FEAT);
    unsigned cstep = (unsigned)((stride * 8ull) % FEAT);

    for (unsigned long long p = tid; p < npair; p += stride) {
        // Speculative prefetch of the next grid-stride unit (OOB is silently
        // dropped per ISA 10.5, so no guard branch needed).
        __builtin_prefetch((const void*)(in4 + 2ull * (p + stride)), 0, 0);

        f32x4 va = __builtin_nontemporal_load(in4 + 2ull * p);
        f32x4 vb = __builtin_nontemporal_load(in4 + 2ull * p + 1ull);

        unsigned c0 = c;
        unsigned c1 = c + 1u; if (c1 >= FEAT) c1 -= FEAT;
        unsigned c2 = c + 2u; if (c2 >= FEAT) c2 -= FEAT;
        unsigned c3 = c + 3u; if (c3 >= FEAT) c3 -= FEAT;
        unsigned c4 = c + 4u; if (c4 >= FEAT) c4 -= FEAT;
        unsigned c5 = c + 5u; if (c5 >= FEAT) c5 -= FEAT;
        unsigned c6 = c + 6u; if (c6 >= FEAT) c6 -= FEAT;
        unsigned c7 = c + 7u; if (c7 >= FEAT) c7 -= FEAT;

        va.x *= mlut[c0];
        va.y *= mlut[c1];
        va.z *= mlut[c2];
        va.w *= mlut[c3];
        vb.x *= mlut[c4];
        vb.y *= mlut[c5];
        vb.z *= mlut[c6];
        vb.w *= mlut[c7];

        __builtin_nontemporal_store(va, out4 + 2ull * p);
        __builtin_nontemporal_store(vb, out4 + 2ull * p + 1ull);

        c += cstep; if (c >= FEAT) c -= FEAT;
    }

    // Scalar-safe tail for elements not covered by full 8-element units.
    // (n = 32768*4337 is divisible by 8, so this is normally empty.)
    unsigned long long rem_base = npair * 8ull;
    for (unsigned long long j = rem_base + tid; j < n; j += stride) {
        unsigned cc = (unsigned)(j % FEAT);
        out[j] = in[j] * mlut[cc];
    }
}

extern "C" void kernel_launch(void* const* d_in, const int* in_sizes, int n_in,
                              void* d_out, int out_size, void* d_ws, size_t ws_size,
                              hipStream_t stream) {
    (void)n_in; (void)d_ws; (void)ws_size; (void)out_size;

    const float* in = (const float*)d_in[0];
    float* out      = (float*)d_out;

    unsigned long long n     = (unsigned long long)in_sizes[0]; // 32768 * 4337
    unsigned long long npair = n / 8ull;

    const int block = 256;                 // 8 wave32 waves per block
    int grid = 8192;                       // ~2M threads, ~8-9 iters each
    unsigned long long needed = (npair + (unsigned long long)block - 1ull) / (unsigned long long)block;
    if (needed == 0ull) needed = 1ull;
    if ((unsigned long long)grid > needed) grid = (int)needed;

    normalize_pixel_pts_kernel<<<grid, block, 0, stream>>>(in, out, npair, n);
}